// BetaTCVAELoss_58566174048495
// MI455X (gfx1250) — compile-verified
//
#include <hip/hip_runtime.h>
#include <hip/hip_bf16.h>
#include <math.h>

typedef __attribute__((ext_vector_type(2))) float v2f;
typedef __attribute__((ext_vector_type(8))) float v8f;
typedef __attribute__((ext_vector_type(4))) int   v4i;

#define B_N 4096
#define L_N 16
#define LOG_2PI 1.8378770664093453f
#define W_TC 2.0f

// ---------------------------------------------------------------------------
// Workspace layout (bytes):
//   F    float4[4096*16]        @ 0        (1 MB)   {u,v,c,0} per (j,l)
//   Bpk  float2[256*16*32]      @ 1 MB     (1 MB)   WMMA B-operand lane layout
//   P1   float [4096*16]        @ 2 MB     (256 KB) sum_j exp(d[i,j,l])
//   P2   float [8*4096]         @ 2.25 MB  (128 KB) partial sum_j exp(S[i,j])
// ---------------------------------------------------------------------------

#if defined(__has_builtin)
#  if __has_builtin(__builtin_amdgcn_global_load_async_to_lds_b128)
#    define HAVE_ASYNC_LDS 1
#  endif
#  if __has_builtin(__builtin_amdgcn_s_wait_asynccnt)
#    define HAVE_WAIT_ASYNC 1
#  endif
#endif

typedef __attribute__((address_space(1))) v4i* gptr_v4i;
typedef __attribute__((address_space(3))) v4i* lptr_v4i;

// Async global->LDS B128 copy (ASYNCcnt path); synchronous fallback otherwise.
__device__ __forceinline__ void async_ld_b128(const float4* gsrc, float4* ldst) {
#if defined(HAVE_ASYNC_LDS)
  __builtin_amdgcn_global_load_async_to_lds_b128(
      (gptr_v4i)gsrc, (lptr_v4i)ldst, /*offset=*/0, /*cpol=*/0);
#else
  *ldst = *gsrc;
#endif
}

__device__ __forceinline__ void wait_async_all() {
#if defined(HAVE_ASYNC_LDS)
#  if defined(HAVE_WAIT_ASYNC)
  __builtin_amdgcn_s_wait_asynccnt(0);
#  else
  asm volatile("s_wait_asynccnt 0x0" ::: "memory");
#  endif
#endif
}

// Kernel 0: per-(x,l) feature computation + packing.
//   u = -0.5*exp(-lv), v = m*exp(-lv), c = -0.5*(m*m*exp(-lv) + lv + LOG_2PI)
// Bpk layout per (j-tile t, l): 32 lanes of float2:
//   lane n   (n<16): {u[16t+n,l], v[16t+n,l]}   -> B VGPR0/1 lanes0-15 (K=0,K=1)
//   lane 16+n      : {c[16t+n,l], 0}            -> B VGPR0/1 lanes16-31 (K=2,K=3)
__global__ __launch_bounds__(256) void prep_kernel(const float* __restrict__ z,
                                                   const float* __restrict__ zm,
                                                   const float* __restrict__ zlv,
                                                   float4* __restrict__ F,
                                                   float2* __restrict__ Bpk) {
  int gid = blockIdx.x * blockDim.x + threadIdx.x;   // 0..65535, = x*16 + l
  int l = gid & 15;
  int x = gid >> 4;
  float lv = zlv[gid];
  float w  = __expf(-lv);
  float u  = -0.5f * w;
  float mv = zm[gid];
  float v  = mv * w;
  float c  = -0.5f * (mv * mv * w + lv + LOG_2PI);
  F[gid] = make_float4(u, v, c, 0.0f);
  int n = x & 15, t = x >> 4;
  float2* bp = Bpk + (size_t)(t * 16 + l) * 32;
  bp[n]      = make_float2(u, v);
  bp[16 + n] = make_float2(c, 0.0f);
}

// Kernel 1: path-1 (per-(i,l) logsumexp over j). Thread = one (i,l) pair,
// scalar accumulator. Features double-buffered through LDS with async
// global->LDS copies overlapped against the exp-heavy compute phase.
#define TJ 128                         // j's per tile
#define NT (B_N / TJ)                  // 32 tiles
#define LDW (TJ * 16 / 256)            // b128 loads per thread per tile = 8
__global__ __launch_bounds__(256) void path1_kernel(const float* __restrict__ z,
                                                    const float4* __restrict__ F,
                                                    float* __restrict__ P1) {
  __shared__ float4 tile[2][TJ * 16];  // 2 x 32 KB of 320 KB WGP LDS
  int tid = threadIdx.x;
  int l  = tid & 15;
  int ii = tid >> 4;
  int i  = blockIdx.x * 16 + ii;
  float zi = z[i * 16 + l];
  float z2 = zi * zi;
  float acc = 0.0f;

  // prologue: fill buffer 0
#pragma unroll
  for (int k = 0; k < LDW; ++k)
    async_ld_b128(&F[tid + 256 * k], &tile[0][tid + 256 * k]);
  wait_async_all();
  __syncthreads();

  for (int jt = 0; jt < NT; ++jt) {
    int cur = jt & 1;
    if (jt + 1 < NT) {                 // prefetch next tile into other buffer
      const float4* src = F + (size_t)(jt + 1) * (TJ * 16);
      float4* dst = tile[(jt + 1) & 1];
#pragma unroll
      for (int k = 0; k < LDW; ++k)
        async_ld_b128(&src[tid + 256 * k], &dst[tid + 256 * k]);
    }
#pragma unroll 8
    for (int jj = 0; jj < TJ; ++jj) {
      float4 f = tile[cur][jj * 16 + l];                 // broadcast, no conflicts
      float d = fmaf(z2, f.x, fmaf(zi, f.y, f.z));
      acc += __expf(d);                                  // d <= ~1.5 -> no shift needed
    }
    wait_async_all();
    __syncthreads();
  }
  P1[i * 16 + l] = acc;
}

// Kernel 2: path-2 via V_WMMA_F32_16X16X4_F32.  One wave per (i-tile r, j-chunk ch).
// S[i,j] = sum_l (z2*u + z*v + c) accumulated across 16 K=4 WMMAs per 16x16 tile
// (two interleaved C accumulators for WMMA ILP), then exp + cross-lane reduce.
__global__ __launch_bounds__(32) void path2_kernel(const float* __restrict__ z,
                                                   const float2* __restrict__ Bpk,
                                                   float* __restrict__ P2) {
  int bx = blockIdx.x;          // 0..2047
  int r  = bx >> 3;             // i-tile: rows 16r..16r+15
  int ch = bx & 7;              // j chunk: tiles [32*ch, 32*ch+32)
  int lane = threadIdx.x;
  int m = lane & 15;

  // A operands (16x4 f32 layout): VGPR0 = {K0:z2 | K2:1}, VGPR1 = {K1:z | K3:0}
  float a0[16], a1[16];
#pragma unroll
  for (int l = 0; l < 16; ++l) {
    float zz = z[(r * 16 + m) * 16 + l];
    a0[l] = (lane < 16) ? zz * zz : 1.0f;
    a1[l] = (lane < 16) ? zz      : 0.0f;
  }

  float part[8] = {0.f, 0.f, 0.f, 0.f, 0.f, 0.f, 0.f, 0.f};
  for (int t = ch * 32; t < ch * 32 + 32; ++t) {
    v8f c0 = {0.f, 0.f, 0.f, 0.f, 0.f, 0.f, 0.f, 0.f};
    v8f c1 = {0.f, 0.f, 0.f, 0.f, 0.f, 0.f, 0.f, 0.f};
    const float2* bt = Bpk + (size_t)t * 16 * 32 + lane;
#pragma unroll
    for (int l = 0; l < 16; l += 2) {
      float2 be = bt[(l + 0) * 32];
      float2 bo = bt[(l + 1) * 32];
      v2f Ae = {a0[l + 0], a1[l + 0]};
      v2f Ao = {a0[l + 1], a1[l + 1]};
      v2f Be = {be.x, be.y};
      v2f Bo = {bo.x, bo.y};
      c0 = __builtin_amdgcn_wmma_f32_16x16x4_f32(false, Ae, false, Be,
                                                 (short)0, c0, false, false);
      c1 = __builtin_amdgcn_wmma_f32_16x16x4_f32(false, Ao, false, Bo,
                                                 (short)0, c1, false, false);
    }
#pragma unroll
    for (int q = 0; q < 8; ++q)
      part[q] += __expf(c0[q] + c1[q]);   // S <= ~25 -> exp fits fp32, no shift
  }

  // reduce over the 16 N-lanes within each half (j dimension)
#pragma unroll
  for (int q = 0; q < 8; ++q) {
    part[q] += __shfl_xor(part[q], 1, 32);
    part[q] += __shfl_xor(part[q], 2, 32);
    part[q] += __shfl_xor(part[q], 4, 32);
    part[q] += __shfl_xor(part[q], 8, 32);
  }
  if (m == 0) {
    int half = lane >> 4;    // D layout: VGPR q holds M=q (lanes 0-15), M=q+8 (16-31)
#pragma unroll
    for (int q = 0; q < 8; ++q)
      P2[ch * B_N + r * 16 + half * 8 + q] = part[q];
  }
}

// Kernel 3: deterministic final reduction -> scalar loss.
__global__ __launch_bounds__(256) void final_kernel(const float* __restrict__ P1,
                                                    const float* __restrict__ P2,
                                                    float* __restrict__ out) {
  __shared__ float red[256];
  int tid = threadIdx.x;
  float local = 0.0f;
  for (int i = tid; i < B_N; i += 256) {
    float s2 = 0.0f;
#pragma unroll
    for (int ch = 0; ch < 8; ++ch) s2 += P2[ch * B_N + i];
    float log_qz = __logf(s2);
    float log_qz_prod = 0.0f;
#pragma unroll
    for (int l = 0; l < 16; ++l) log_qz_prod += __logf(P1[i * 16 + l]);
    local += (log_qz - log_qz_prod);
  }
  red[tid] = local;
  __syncthreads();
  for (int s = 128; s > 0; s >>= 1) {
    if (tid < s) red[tid] += red[tid + s];
    __syncthreads();
  }
  if (tid == 0) out[0] = (W_TC - 1.0f) * red[0] / (float)B_N;
}

extern "C" void kernel_launch(void* const* d_in, const int* in_sizes, int n_in,
                              void* d_out, int out_size, void* d_ws, size_t ws_size,
                              hipStream_t stream) {
  (void)in_sizes; (void)n_in; (void)out_size; (void)ws_size;
  const float* z   = (const float*)d_in[0];
  const float* zm  = (const float*)d_in[1];
  const float* zlv = (const float*)d_in[2];
  char* ws = (char*)d_ws;
  float4* F   = (float4*)(ws);
  float2* Bpk = (float2*)(ws + (1u << 20));
  float*  P1  = (float*)(ws + (2u << 20));
  float*  P2  = (float*)(ws + (2u << 20) + (256u << 10));

  prep_kernel <<<256, 256, 0, stream>>>(z, zm, zlv, F, Bpk);
  path1_kernel<<<256, 256, 0, stream>>>(z, F, P1);
  path2_kernel<<<2048, 32, 0, stream>>>(z, Bpk, P2);
  final_kernel<<<1, 256, 0, stream>>>(P1, P2, (float*)d_out);
}